// RPNModule_43808666419879
// MI455X (gfx1250) — compile-verified
//
#include <hip/hip_runtime.h>
#include <hip/hip_bf16.h>

// ---------------------------------------------------------------------------
// RPN forward for MI455X (gfx1250, wave32, WMMA).
// Conv 3x3 + heads on V_WMMA_F32_16X16X4_F32 (full fp32 matrix path), with
// async global->LDS staging of activations and conflict-free ds_load_b64
// fragment reads. Top-k via threshold bisection + LDS bitonic sort; NMS in LDS.
// ---------------------------------------------------------------------------

typedef float v2f __attribute__((ext_vector_type(2)));
typedef float v8f __attribute__((ext_vector_type(8)));

#define BB   2
#define CC_  256
#define HH   64
#define WW   64
#define AA   15
#define NANCH (HH*WW*AA)          // 61440 per image
#define PRE_NMS  6000
#define POST_NMS 1000
#define CAND 8192                 // sort capacity (LDS)
#define KCONV (9*256)             // 2304
#define NMS_THRESH 0.7f
#define BBOX_CLIP 4.135166556742356f  // log(1000/16)

#define PIX_STRIDE 260            // padded LDS pixel stride (floats): bank-conflict-free b64

// ---------------- workspace layout (float units) ----------------
#define OFF_WT      ((size_t)0)                       // 2304*256 paired layout
#define OFF_WT2     (OFF_WT + (size_t)KCONV*256)      // 256*80 paired layout
#define OFF_T       (OFF_WT2 + (size_t)256*80)        // 2*256*4096
#define OFF_LOGITS  (OFF_T + (size_t)BB*CC_*HH*WW)    // 2*61440
#define OFF_DELTAS  (OFF_LOGITS + (size_t)BB*NANCH)   // 2*61440*4
#define OFF_SELIDX  (OFF_DELTAS + (size_t)BB*NANCH*4) // 2*6000 (int)
#define OFF_SELLOG  (OFF_SELIDX + (size_t)BB*PRE_NMS)
#define OFF_BOXES   (OFF_SELLOG + (size_t)BB*PRE_NMS) // 2*6000*4
#define OFF_SCORES  (OFF_BOXES + (size_t)BB*PRE_NMS*4)
#define OFF_VALID   (OFF_SCORES + (size_t)BB*PRE_NMS) // 2*6000 (int)

__device__ __forceinline__ unsigned lds_addr32(const void* p) {
  // ISA 10.2: flat->LDS aperture uses addr[31:0] as the LDS address.
  return (unsigned)(unsigned long long)(uintptr_t)p;
}

__device__ __forceinline__ void async_g2l_b32(unsigned lds_off,
                                              unsigned long long gaddr) {
  asm volatile("global_load_async_to_lds_b32 %0, %1, off"
               :: "v"(lds_off), "v"(gaddr) : "memory");
}

__device__ __forceinline__ void wait_async0() {
  asm volatile("s_wait_asynccnt 0x0" ::: "memory");
}

// ---------------------------------------------------------------------------
// Stage 0: weight transforms into pair-interleaved GEMM layouts:
//   wTp [((tap*128 + c/2)*256 + o)*2 + (c&1)] = conv_w[o][c][tap]   (b64 B-frags)
//   wT2p[(( c/2)*80 + n)*2 + (c&1)]           = head_w[n][c]        (n<15 obj, <75 bbox)
// ---------------------------------------------------------------------------
__global__ void prep_weights(const float* __restrict__ conv_w,
                             const float* __restrict__ obj_w,
                             const float* __restrict__ bbox_w,
                             float* __restrict__ wTp,
                             float* __restrict__ wT2p) {
  int t = blockIdx.x * blockDim.x + threadIdx.x;
  const int N1 = KCONV * 256;
  if (t < N1) {
    int k = t >> 8;          // tap*256 + c
    int o = t & 255;         // out channel
    int tap = k >> 8;
    int c   = k & 255;
    size_t dst = ((size_t)(tap * 128 + (c >> 1)) * 256 + o) * 2 + (c & 1);
    wTp[dst] = conv_w[o * 2304 + c * 9 + tap];
  } else if (t < N1 + 256 * 80) {
    int u = t - N1;
    int c = u / 80, n = u % 80;
    float v = 0.f;
    if (n < 15)      v = obj_w[n * 256 + c];
    else if (n < 75) v = bbox_w[(n - 15) * 256 + c];
    size_t dst = ((size_t)(c >> 1) * 80 + n) * 2 + (c & 1);
    wT2p[dst] = v;
  }
}

// ---------------------------------------------------------------------------
// Stage 1: 3x3 conv + bias + ReLU, implicit GEMM on V_WMMA_F32_16X16X4_F32.
// Block = 4 waves, grid (512 M-tiles, 2). Wave owns TWO 16x16 N tiles sharing
// each A fragment (2 wmma / A-load, 2 independent chains). The 3x18x256 halo
// tile is async-staged into LDS [pixel][channel] (stride 260 -> conflict-free
// ds_load_b64: lanes 0-15 banks {4l,4l+1}, lanes 16-31 banks {4l+2,4l+3}).
// ---------------------------------------------------------------------------
__global__ __launch_bounds__(128) void conv3x3_wmma(
    const float* __restrict__ feat, const float* __restrict__ wTp,
    const float* __restrict__ conv_b, float* __restrict__ tout_base) {
  extern __shared__ float sA[];            // 54 * 260 floats = 56160 B
  const int tid  = threadIdx.x;
  const int wave = tid >> 5;
  const int lane = tid & 31;
  const int half = lane >> 4;
  const int l    = lane & 15;

  const int bx = blockIdx.x;               // 512 M tiles
  const int b  = bx >> 8;
  const int y  = (bx >> 2) & 63;
  const int x0 = (bx & 3) << 4;
  const int n0 = blockIdx.y * 128 + wave * 32;   // tiles n0 and n0+16

  const float* fb = feat + (size_t)b * (CC_ * HH * WW);

  // ---- async cooperative fill: i = c*54 + rr*18 + xx (xx fastest: coalesced)
  for (int i = tid; i < 3 * 18 * 256; i += 128) {
    const int xx = i % 18;
    const int rr = (i / 18) % 3;
    const int c  = i / 54;
    const int yy = y + rr - 1;
    const int gx = x0 + xx - 1;
    float* dst = sA + (rr * 18 + xx) * PIX_STRIDE + c;
    if (yy >= 0 && yy < HH && gx >= 0 && gx < WW) {
      async_g2l_b32(lds_addr32(dst),
                    (unsigned long long)(uintptr_t)(fb + (size_t)c * 4096 + yy * 64 + gx));
    } else {
      *dst = 0.f;
    }
  }
  wait_async0();
  __syncthreads();

  v8f acc0 = {0.f, 0.f, 0.f, 0.f, 0.f, 0.f, 0.f, 0.f};
  v8f acc1 = {0.f, 0.f, 0.f, 0.f, 0.f, 0.f, 0.f, 0.f};

  for (int tap = 0; tap < 9; ++tap) {
    const int s = tap % 3;
    const int r = tap / 3;
    const float* aRow = sA + (r * 18 + l + s) * PIX_STRIDE;   // + cc
    const float* wRow = wTp + (size_t)(tap * 128) * 256 * 2;  // + (cgrp*256+n)*2
    __builtin_prefetch(wRow + (size_t)(n0 + l) * 2, 0, 1);    // global_prefetch_b8
    for (int c = 0; c < 256; c += 4) {
      const int cc = c + 2 * half;                            // even
      const v2f a  = *(const v2f*)(aRow + cc);                // ds_load_b64
      const v2f w0 = *(const v2f*)(wRow + ((size_t)(cc >> 1) * 256 + n0 + l) * 2);
      const v2f w1 = *(const v2f*)(wRow + ((size_t)(cc >> 1) * 256 + n0 + 16 + l) * 2);
      acc0 = __builtin_amdgcn_wmma_f32_16x16x4_f32(
          false, a, false, w0, (short)0, acc0, false, false);
      acc1 = __builtin_amdgcn_wmma_f32_16x16x4_f32(
          false, a, false, w1, (short)0, acc1, false, false);
    }
  }

  // epilogue: D layout -> lanes 0-15: M=v, lanes 16-31: M=v+8; N = ntile + l.
#pragma unroll
  for (int t2 = 0; t2 < 2; ++t2) {
    const int n = n0 + 16 * t2 + l;
    const float bias = conv_b[n];
    float* trow = tout_base + ((size_t)(b * CC_ + n)) * (HH * WW) + y * WW;
    const v8f acc = t2 ? acc1 : acc0;
#pragma unroll
    for (int v = 0; v < 8; ++v) {
      const int row = v + 8 * half;
      float val = acc[v] + bias;
      trow[x0 + row] = val > 0.f ? val : 0.f;
    }
  }
}

// ---------------------------------------------------------------------------
// Stage 2: heads GEMM M=8192, N=80 (15 obj + 60 bbox + pad), K=256.
// Activation tile (16 pix x 256 ch) async-staged into LDS, b64 frags.
// ---------------------------------------------------------------------------
__global__ __launch_bounds__(160) void head_wmma(
    const float* __restrict__ tin, const float* __restrict__ wT2p,
    const float* __restrict__ obj_b, const float* __restrict__ bbox_b,
    float* __restrict__ logits, float* __restrict__ deltas) {
  extern __shared__ float sT[];            // 16 * 260 floats = 16640 B
  const int tid  = threadIdx.x;
  const int wave = tid >> 5;
  const int lane = tid & 31;
  const int half = lane >> 4;
  const int l    = lane & 15;

  const int bx = blockIdx.x;
  const int b  = bx >> 8;
  const int y  = (bx >> 2) & 63;
  const int x0 = (bx & 3) << 4;
  const int n0 = wave * 16;

  const float* tb = tin + (size_t)b * (CC_ * HH * WW) + y * WW;

  // async fill: i = c*16 + xx (xx fastest: coalesced 64B runs)
  for (int i = tid; i < 16 * 256; i += 160) {
    const int xx = i & 15;
    const int c  = i >> 4;
    float* dst = sT + xx * PIX_STRIDE + c;
    async_g2l_b32(lds_addr32(dst),
                  (unsigned long long)(uintptr_t)(tb + (size_t)c * 4096 + x0 + xx));
  }
  wait_async0();
  __syncthreads();

  v8f acc = {0.f, 0.f, 0.f, 0.f, 0.f, 0.f, 0.f, 0.f};
  const float* aRow = sT + l * PIX_STRIDE;
  for (int c = 0; c < 256; c += 4) {
    const int cc = c + 2 * half;
    const v2f a = *(const v2f*)(aRow + cc);
    const v2f w = *(const v2f*)(wT2p + ((size_t)(cc >> 1) * 80 + n0 + l) * 2);
    acc = __builtin_amdgcn_wmma_f32_16x16x4_f32(
        false, a, false, w, (short)0, acc, false, false);
  }

  const int n = n0 + l;
  if (n < 75) {
    const float bias = (n < 15) ? obj_b[n] : bbox_b[n - 15];
#pragma unroll
    for (int v = 0; v < 8; ++v) {
      const int row = v + 8 * half;
      const int x = x0 + row;
      const int cell = y * WW + x;
      const float val = acc[v] + bias;
      if (n < 15) {
        logits[(size_t)b * NANCH + cell * AA + n] = val;
      } else {
        const int q = n - 15, an = q >> 2, comp = q & 3;
        deltas[((size_t)b * NANCH + (size_t)cell * AA + an) * 4 + comp] = val;
      }
    }
  }
}

// ---------------------------------------------------------------------------
// Stage 3: top-6000 per image: threshold bisection + LDS bitonic sort of 8192.
// ---------------------------------------------------------------------------
__global__ void topk_select(const float* __restrict__ logits,
                            int* __restrict__ sel_idx,
                            float* __restrict__ sel_logit) {
  const int b = blockIdx.x;
  const float* L = logits + (size_t)b * NANCH;
  extern __shared__ char smem[];
  float* sscore = (float*)smem;                 // CAND floats
  int*   sidx   = (int*)(smem + CAND * 4);      // CAND ints
  __shared__ float red_f[1024];
  __shared__ int   red_i[1024];
  __shared__ int   s_cnt;

  const int tid = threadIdx.x, nt = blockDim.x;

  float mx = -1e30f, mn = 1e30f;
  for (int i = tid; i < NANCH; i += nt) {
    float v = L[i];
    mx = fmaxf(mx, v);
    mn = fminf(mn, v);
  }
  red_f[tid] = mx; __syncthreads();
  for (int s = nt >> 1; s > 0; s >>= 1) {
    if (tid < s) red_f[tid] = fmaxf(red_f[tid], red_f[tid + s]);
    __syncthreads();
  }
  float hi = red_f[0] + 1.0f; __syncthreads();
  red_f[tid] = mn; __syncthreads();
  for (int s = nt >> 1; s > 0; s >>= 1) {
    if (tid < s) red_f[tid] = fminf(red_f[tid], red_f[tid + s]);
    __syncthreads();
  }
  float lo = red_f[0] - 1.0f; __syncthreads();

  for (int it = 0; it < 48; ++it) {
    const float mid = 0.5f * (lo + hi);
    int c = 0;
    for (int i = tid; i < NANCH; i += nt) c += (L[i] > mid) ? 1 : 0;
    red_i[tid] = c; __syncthreads();
    for (int s = nt >> 1; s > 0; s >>= 1) {
      if (tid < s) red_i[tid] += red_i[tid + s];
      __syncthreads();
    }
    const int total = red_i[0];
    __syncthreads();
    if (total > PRE_NMS) lo = mid; else hi = mid;
  }

  for (int i = tid; i < CAND; i += nt) { sscore[i] = -1e30f; sidx[i] = 0; }
  if (tid == 0) s_cnt = 0;
  __syncthreads();
  for (int i = tid; i < NANCH; i += nt) {
    const float v = L[i];
    if (v > lo) {
      const int p = atomicAdd(&s_cnt, 1);
      if (p < CAND) { sscore[p] = v; sidx[p] = i; }
    }
  }
  __syncthreads();

  for (int k = 2; k <= CAND; k <<= 1) {
    for (int j = k >> 1; j > 0; j >>= 1) {
      for (int i = tid; i < CAND; i += nt) {
        const int ixj = i ^ j;
        if (ixj > i) {
          const bool desc = ((i & k) == 0);
          const float a = sscore[i], c = sscore[ixj];
          const bool sw = desc ? (a < c) : (a > c);
          if (sw) {
            sscore[i] = c; sscore[ixj] = a;
            const int ti = sidx[i]; sidx[i] = sidx[ixj]; sidx[ixj] = ti;
          }
        }
      }
      __syncthreads();
    }
  }

  for (int i = tid; i < PRE_NMS; i += nt) {
    sel_idx[(size_t)b * PRE_NMS + i]   = sidx[i];
    sel_logit[(size_t)b * PRE_NMS + i] = sscore[i];
  }
}

// ---------------------------------------------------------------------------
// Stage 4: decode + clip + sigmoid + validity.
// ---------------------------------------------------------------------------
__global__ void decode_kernel(const int* __restrict__ sel_idx,
                              const float* __restrict__ sel_logit,
                              const float* __restrict__ deltas,
                              float* __restrict__ boxes,
                              float* __restrict__ scores,
                              int* __restrict__ valid) {
  const int t = blockIdx.x * blockDim.x + threadIdx.x;
  if (t >= BB * PRE_NMS) return;
  const int b = t / PRE_NMS;
  const int idx = sel_idx[t];

  const int a = idx % AA;
  const int cell = idx / AA;
  const int ay = cell >> 6, ax = cell & 63;
  const float sizes[5]  = {32.f, 64.f, 128.f, 256.f, 512.f};
  const float ratios[3] = {0.5f, 1.f, 2.f};
  const float S = sizes[a / 3], R = ratios[a % 3];
  const float w0 = sqrtf(S * S / R);
  const float h0 = w0 * R;
  const float sx = ax * 16.f, sy = ay * 16.f;
  const float ax1 = sx - (w0 - 1.f) * 0.5f, ay1 = sy - (h0 - 1.f) * 0.5f;
  const float ax2 = sx + (w0 - 1.f) * 0.5f, ay2 = sy + (h0 - 1.f) * 0.5f;
  const float aw = ax2 - ax1 + 1.f, ah = ay2 - ay1 + 1.f;
  const float acx = ax1 + 0.5f * aw, acy = ay1 + 0.5f * ah;

  const float* d = deltas + ((size_t)b * NANCH + idx) * 4;
  const float dx = d[0], dy = d[1];
  const float dw = fminf(d[2], BBOX_CLIP);
  const float dh = fminf(d[3], BBOX_CLIP);
  const float pcx = dx * aw + acx, pcy = dy * ah + acy;
  const float pw = __expf(dw) * aw, ph = __expf(dh) * ah;

  float x1 = pcx - 0.5f * pw, y1 = pcy - 0.5f * ph;
  float x2 = pcx + 0.5f * pw - 1.f, y2 = pcy + 0.5f * ph - 1.f;
  x1 = fminf(fmaxf(x1, 0.f), 1023.f);
  y1 = fminf(fmaxf(y1, 0.f), 1023.f);
  x2 = fminf(fmaxf(x2, 0.f), 1023.f);
  y2 = fminf(fmaxf(y2, 0.f), 1023.f);

  float* bo = boxes + (size_t)t * 4;
  bo[0] = x1; bo[1] = y1; bo[2] = x2; bo[3] = y2;
  scores[t] = 1.f / (1.f + __expf(-sel_logit[t]));
  valid[t] = ((x2 - x1 + 1.f) >= 0.f) && ((y2 - y1 + 1.f) >= 0.f);
}

// ---------------------------------------------------------------------------
// Stage 5: greedy NMS (all state in 144 KB LDS) + in-order top-1000 emit.
// ---------------------------------------------------------------------------
__global__ void nms_kernel(const float* __restrict__ boxes,
                           const float* __restrict__ scores,
                           const int* __restrict__ valid,
                           float* __restrict__ out) {
  const int b = blockIdx.x;
  const int tid = threadIdx.x, nt = blockDim.x;
  extern __shared__ char smem[];
  float4* sb    = (float4*)smem;                               // 96000 B
  float*  sarea = (float*)(smem + PRE_NMS * 16);               // 24000 B
  int*    ssup  = (int*)(smem + PRE_NMS * 16 + PRE_NMS * 4);   // 24000 B

  const float4* Bx = (const float4*)(boxes + (size_t)b * PRE_NMS * 4);
  for (int i = tid; i < PRE_NMS; i += nt) {
    const float4 v = Bx[i];
    sb[i] = v;
    sarea[i] = (v.z - v.x + 1.f) * (v.w - v.y + 1.f);
    ssup[i] = valid[(size_t)b * PRE_NMS + i] ? 0 : 1;
  }
  __syncthreads();

  for (int i = 0; i < PRE_NMS; ++i) {
    if (!ssup[i]) {   // uniform shared value -> no divergence at the barrier
      const float4 bi = sb[i];
      const float ai = sarea[i];
      for (int j = i + 1 + tid; j < PRE_NMS; j += nt) {
        if (!ssup[j]) {
          const float xx1 = fmaxf(bi.x, sb[j].x);
          const float yy1 = fmaxf(bi.y, sb[j].y);
          const float xx2 = fminf(bi.z, sb[j].z);
          const float yy2 = fminf(bi.w, sb[j].w);
          const float w = fmaxf(xx2 - xx1 + 1.f, 0.f);
          const float h = fmaxf(yy2 - yy1 + 1.f, 0.f);
          const float inter = w * h;
          const float iou = inter / (ai + sarea[j] - inter);
          if (iou > NMS_THRESH) ssup[j] = 1;
        }
      }
    }
    __syncthreads();
  }

  if (tid == 0) {
    float* O = out + (size_t)b * POST_NMS * 5;
    int cnt = 0;
    for (int i = 0; i < PRE_NMS && cnt < POST_NMS; ++i) {
      if (!ssup[i]) {
        O[cnt * 5 + 0] = sb[i].x;
        O[cnt * 5 + 1] = sb[i].y;
        O[cnt * 5 + 2] = sb[i].z;
        O[cnt * 5 + 3] = sb[i].w;
        O[cnt * 5 + 4] = scores[(size_t)b * PRE_NMS + i];
        ++cnt;
      }
    }
    for (; cnt < POST_NMS; ++cnt)
      for (int k = 0; k < 5; ++k) O[cnt * 5 + k] = 0.f;
  }
}

// ---------------------------------------------------------------------------
extern "C" void kernel_launch(void* const* d_in, const int* in_sizes, int n_in,
                              void* d_out, int out_size, void* d_ws, size_t ws_size,
                              hipStream_t stream) {
  const float* features = (const float*)d_in[0];
  const float* conv_w   = (const float*)d_in[1];
  const float* conv_b   = (const float*)d_in[2];
  const float* obj_w    = (const float*)d_in[3];
  const float* obj_b    = (const float*)d_in[4];
  const float* bbox_w   = (const float*)d_in[5];
  const float* bbox_b   = (const float*)d_in[6];

  float* ws = (float*)d_ws;
  float* wTp     = ws + OFF_WT;
  float* wT2p    = ws + OFF_WT2;
  float* tbuf    = ws + OFF_T;
  float* logits  = ws + OFF_LOGITS;
  float* deltas  = ws + OFF_DELTAS;
  int*   selidx  = (int*)(ws + OFF_SELIDX);
  float* sellog  = ws + OFF_SELLOG;
  float* boxes   = ws + OFF_BOXES;
  float* scores  = ws + OFF_SCORES;
  int*   valid   = (int*)(ws + OFF_VALID);

  {
    const int n = KCONV * 256 + 256 * 80;
    prep_weights<<<(n + 255) / 256, 256, 0, stream>>>(conv_w, obj_w, bbox_w, wTp, wT2p);
  }
  conv3x3_wmma<<<dim3(512, 2), 128, 54 * PIX_STRIDE * 4, stream>>>(
      features, wTp, conv_b, tbuf);
  head_wmma<<<512, 160, 16 * PIX_STRIDE * 4, stream>>>(
      tbuf, wT2p, obj_b, bbox_b, logits, deltas);
  topk_select<<<BB, 1024, CAND * 8, stream>>>(logits, selidx, sellog);
  decode_kernel<<<(BB * PRE_NMS + 255) / 256, 256, 0, stream>>>(
      selidx, sellog, deltas, boxes, scores, valid);
  nms_kernel<<<BB, 256, PRE_NMS * 24, stream>>>(boxes, scores, valid, (float*)d_out);
}